// RNNTWordDistillLoss_75548474737116
// MI455X (gfx1250) — compile-verified
//
#include <hip/hip_runtime.h>

typedef __attribute__((ext_vector_type(4))) float v4f;
typedef __attribute__((ext_vector_type(2))) float v2f;
typedef __attribute__((ext_vector_type(8))) float v8f;

#define B_ 8
#define T_ 256
#define U_ 48
#define V_ 1024
#define NROWS (B_ * T_ * U_)          // 98304 rows of V floats
#define WPB 8                          // waves per block (wave32)
#define NBLOCKS (NROWS / WPB)          // 12288 partials (divisible by 64*32)

__device__ __forceinline__ float wave_reduce_add(float v) {
#pragma unroll
    for (int off = 16; off > 0; off >>= 1) v += __shfl_xor(v, off, 32);
    return v;
}
__device__ __forceinline__ float wave_reduce_max(float v) {
#pragma unroll
    for (int off = 16; off > 0; off >>= 1) v = fmaxf(v, __shfl_xor(v, off, 32));
    return v;
}

// One wave32 per (b,t,u) row: 32 lanes x 32 floats = V=1024.
// Streams logits with non-temporal b128 loads (read-once, 402MB > L2: keep the
// 1.5MB soft_labels resident instead), keeps the row in registers, does
// max / dot / sum-exp with wave32 shfl ladders. Masked rows are skipped by a
// wave-uniform branch so their 4KB is never fetched (~40% traffic saved).
__global__ __launch_bounds__(256) void rnnt_row_kernel(
    const float* __restrict__ logits, const float* __restrict__ soft,
    const int* __restrict__ xlens, const int* __restrict__ ylens,
    float* __restrict__ partial) {
    __shared__ float sred[WPB];
    const int tid  = threadIdx.x;
    const int lane = tid & 31;
    const int wave = tid >> 5;
    const int r  = blockIdx.x * WPB + wave;      // row id = ((b*T+t)*U+u)
    const int u  = r % U_;
    const int bt = r / U_;
    const int t  = bt % T_;
    const int b  = bt / T_;
    const int xl = xlens[b];
    const int yl = ylens[b];

    float val = 0.0f;
    if (t < xl && u < yl) {                      // wave-uniform mask
        const float* xrow = logits + (size_t)r * V_;
        const float* prow = soft + ((size_t)b * U_ + u) * (size_t)V_;
        v4f xr[8];
        float dot = 0.0f;
        float m   = -3.402823466e+38f;
#pragma unroll
        for (int i = 0; i < 8; ++i) {
            const int off = i * 128 + lane * 4;  // 512B contiguous per wave-load
            v4f xv = __builtin_nontemporal_load((const v4f*)(xrow + off));
            v4f pv = *(const v4f*)(prow + off);  // reused 256x across t: keep cached
            xr[i] = xv;
            dot = fmaf(xv.x, pv.x, dot);
            dot = fmaf(xv.y, pv.y, dot);
            dot = fmaf(xv.z, pv.z, dot);
            dot = fmaf(xv.w, pv.w, dot);
            m = fmaxf(m, fmaxf(fmaxf(xv.x, xv.y), fmaxf(xv.z, xv.w)));
        }
        m   = wave_reduce_max(m);
        dot = wave_reduce_add(dot);
        float s = 0.0f;
#pragma unroll
        for (int i = 0; i < 8; ++i) {
            s += __expf(xr[i].x - m) + __expf(xr[i].y - m) +
                 __expf(xr[i].z - m) + __expf(xr[i].w - m);
        }
        s = wave_reduce_add(s);
        // contrib = dot(p,x) - logsumexp(x); pre-divide by (xl*yl)
        val = (dot - (m + __logf(s))) / (float)(xl * yl);
    }
    if (lane == 0) sred[wave] = val;
    __syncthreads();
    if (tid == 0) {
        float acc = 0.0f;
#pragma unroll
        for (int w = 0; w < WPB; ++w) acc += sred[w];
        partial[blockIdx.x] = acc;
    }
}

// Deterministic final reduction of the 12288 block partials on one workgroup.
// V_WMMA_F32_16X16X4_F32 with B = ones: D[m,n] += sum_k A[m,k], i.e. an exact
// fp32 fixed-order sum tree on the matrix pipe. Each WMMA consumes 64 values
// per wave (2 VGPRs/lane of A). NBLOCKS is divisible by 64*32, so every wave
// runs exactly NBLOCKS/2048 = 6 iterations with no tail predication: the loop
// body is one global_load_b64 + one v_wmma. EXEC is all-ones throughout
// (wave-uniform trip counts, no divergence).
__global__ __launch_bounds__(1024) void rnnt_reduce_kernel(
    const float* __restrict__ part, float* __restrict__ out) {
    __shared__ float sdata[32];
    const int tid  = threadIdx.x;
    const int lane = tid & 31;
    const int wave = tid >> 5;

    v8f acc = {};
    v2f ones;
    ones.x = 1.0f;
    ones.y = 1.0f;

    constexpr int CHUNKS = NBLOCKS / 64;         // 192; 6 per wave exactly
#pragma unroll 2
    for (int c = wave; c < CHUNKS; c += 32) {
        v2f a = *(const v2f*)(part + c * 64 + lane * 2);  // clean b64 load
        acc = __builtin_amdgcn_wmma_f32_16x16x4_f32(
            /*neg_a=*/false, a, /*neg_b=*/false, ones,
            /*c_mod=*/(short)0, acc, /*reuse_a=*/false, /*reuse_b=*/false);
    }
    // Every column of D is identical; total = sum over m of D[m,0].
    // D[m,0]: M=0..7 -> acc[0..7] at lane 0; M=8..15 -> acc[0..7] at lane 16.
    float v = 0.0f;
    if (lane == 0 || lane == 16) {
#pragma unroll
        for (int j = 0; j < 8; ++j) v += acc[j];
    }
    v = wave_reduce_add(v);
    if (lane == 0) sdata[wave] = v;
    __syncthreads();
    if (tid == 0) {
        float tot = 0.0f;
#pragma unroll
        for (int w = 0; w < 32; ++w) tot += sdata[w];
        out[0] = -tot * (1.0f / (float)B_);      // negate + /B
    }
}

extern "C" void kernel_launch(void* const* d_in, const int* in_sizes, int n_in,
                              void* d_out, int out_size, void* d_ws, size_t ws_size,
                              hipStream_t stream) {
    const float* logits = (const float*)d_in[0];   // [B,T,U,V] fp32
    const float* soft   = (const float*)d_in[1];   // [B,U,V]   fp32
    const int*   xlens  = (const int*)d_in[2];     // [B]
    const int*   ylens  = (const int*)d_in[3];     // [B]
    float* partial = (float*)d_ws;                 // NBLOCKS floats (48KB)
    float* out     = (float*)d_out;                // scalar

    rnnt_row_kernel<<<NBLOCKS, 256, 0, stream>>>(logits, soft, xlens, ylens, partial);
    rnnt_reduce_kernel<<<1, 1024, 0, stream>>>(partial, out);
}